// RankingLoss_53034256171624
// MI455X (gfx1250) — compile-verified
//
#include <hip/hip_runtime.h>
#include <hip/hip_bf16.h>

typedef float v2f __attribute__((ext_vector_type(2)));
typedef float v8f __attribute__((ext_vector_type(8)));
typedef unsigned int u32x4 __attribute__((ext_vector_type(4)));
typedef int i32x4 __attribute__((ext_vector_type(4)));
typedef int i32x8 __attribute__((ext_vector_type(8)));

#define BM 128
#define BN 64
#define BK 32
#define LDA 36   // 32 dwords/row + 4 dwords TDM pad -> 16B-aligned rows, conflict-free b64 frag loads
#define LDB 65   // 32x64 B tile, stride 65 floats: conflict-free column reads

// ---------------------------------------------------------------------------
// Kernel 1: gt[b] = dot(pred[b,:], sig[:, label[b]]); also zero the output.
// ---------------------------------------------------------------------------
__global__ __launch_bounds__(256) void rankloss_gt_kernel(
    const float* __restrict__ pred, const int* __restrict__ label,
    const float* __restrict__ sig, float* __restrict__ gt,
    float* __restrict__ out, int D, int C)
{
    if (blockIdx.x == 0 && threadIdx.x == 0) out[0] = 0.0f;
    const int wid  = threadIdx.x >> 5;
    const int lane = threadIdx.x & 31;
    const int b    = blockIdx.x * 8 + wid;
    const int col  = label[b];
    const float* prow = pred + (size_t)b * D;
    float s = 0.0f;
    for (int i = lane; i < D; i += 32)
        s += prow[i] * sig[(size_t)i * C + col];
    #pragma unroll
    for (int off = 16; off > 0; off >>= 1)
        s += __shfl_xor(s, off, 32);
    if (lane == 0) gt[b] = s;
}

// ---------------------------------------------------------------------------
// TDM: DMA a BM x BK f32 tile (row stride D elements) into LDS with hardware
// padding producing a 36-dword LDS row pitch (pad 4 dwords every 32 dwords).
// ---------------------------------------------------------------------------
__device__ __forceinline__ void tdm_load_A_tile(const float* gptr,
                                                unsigned lds_off, int D)
{
    unsigned long long ga = (unsigned long long)(uintptr_t)gptr;
    u32x4 g0;
    g0[0] = 1u;                                   // count=1, load, user D#
    g0[1] = lds_off;                              // lds_addr (bytes)
    g0[2] = (unsigned)ga;                         // global_addr[31:0]
    g0[3] = ((unsigned)(ga >> 32) & 0x01FFFFFFu)  // global_addr[56:32]
          | (2u << 30);                           // type = 2 ("image")

    i32x8 g1;
    g1[0] = (2 << 16)       // data_size = 4 bytes
          | (1 << 20)       // pad_enable
          | (4 << 22)       // pad_interval: after every 32 dwords
          | (3 << 25);      // pad_amount: 4 dwords  => LDS pitch = 36 floats
    g1[1] = (BK & 0xFFFF) << 16;   // tensor_dim0 = 32 (tiles never OOB)
    g1[2] = (BM & 0xFFFF) << 16;   // tensor_dim0[31:16]=0 | tensor_dim1 = 128
    g1[3] = (BK & 0xFFFF) << 16;   // tensor_dim1[31:16]=0 | tile_dim0 = 32
    g1[4] = (BM & 0xFFFF);         // tile_dim1 = 128, tile_dim2 = 0
    g1[5] = D;                     // tensor_dim0_stride = D elements
    g1[6] = 0;
    g1[7] = 0;

    i32x4 z4 = {0, 0, 0, 0};
#if defined(__clang_major__) && (__clang_major__ >= 23)
    i32x8 z8 = {0, 0, 0, 0, 0, 0, 0, 0};
    __builtin_amdgcn_tensor_load_to_lds(g0, g1, z4, z4, z8, 0);
#else
    __builtin_amdgcn_tensor_load_to_lds(g0, g1, z4, z4, 0);
#endif
}

// ---------------------------------------------------------------------------
// Kernel 2: fused GEMM (f32 WMMA 16x16x4) + masked hinge + global reduction.
// Block tile 128x64, 8 waves, each wave -> 32x32 (2x2 WMMA 16x16 tiles).
// A tile arrives via TENSOR_LOAD_TO_LDS (wave 0 issues, TENSORcnt + barrier).
// ---------------------------------------------------------------------------
__global__ __launch_bounds__(256) void rankloss_gemm_kernel(
    const float* __restrict__ pred, const float* __restrict__ sig,
    const int* __restrict__ tcls, const int* __restrict__ label,
    const float* __restrict__ gt, float* __restrict__ out,
    int C, int D)
{
    __shared__ __align__(16) float As[BM * LDA];
    __shared__ __align__(16) float Bs[BK * LDB];
    __shared__ int   tcs[BN];
    __shared__ float red[8];

    const int tid  = threadIdx.x;
    const int wid  = tid >> 5;
    const int lane = tid & 31;
    const int hi   = lane >> 4;   // 0: lanes 0-15, 1: lanes 16-31
    const int ml   = lane & 15;
    const int wm   = wid & 3;     // wave row (4 x 32 rows = 128)
    const int wn   = wid >> 2;    // wave col (2 x 32 cols = 64)

    const int row0 = blockIdx.y * BM;
    const int col0 = blockIdx.x * BN;

    const unsigned As_base = (unsigned)(uintptr_t)(void*)&As[0];

    if (tid < BN) {
        int c = col0 + tid;
        tcs[tid] = (c < C) ? tcls[c] : -1;    // -1 sentinel: never equals a label
    }
    __syncthreads();

    v8f acc[2][2] = {};

    const int bn = tid & 63;          // B tile column
    const int bk = tid >> 6;          // 0..3, B row per pass

    int gc = tcs[bn];
    if (gc < 0 || gc >= C) gc = 0;    // clamp gather (masked out in epilogue)
    const float* sigcol = sig + gc;

    for (int k0 = 0; k0 < D; k0 += BK) {
        // A tile: one TDM descriptor, issued by wave 0 only (EXEC-independent,
        // wave-level DMA; completion fanned out through the workgroup barrier)
        if (wid == 0)
            tdm_load_A_tile(pred + (size_t)row0 * D + k0, As_base, D);

        // B tile: 32x64, per-element column gather through train_classes
        #pragma unroll
        for (int p = 0; p < 8; p++) {
            int k = bk + p * 4;
            Bs[k * LDB + bn] = sigcol[(size_t)(k0 + k) * C];
        }
        if (k0 + BK < D)
            __builtin_prefetch(sigcol + (size_t)(k0 + BK + bk) * C, 0, 0);

        if (wid == 0)
            __builtin_amdgcn_s_wait_tensorcnt(0);
        __syncthreads();

        // 8 K-steps of V_WMMA_F32_16X16X4_F32 per K-chunk, 4 tiles each
        #pragma unroll
        for (int kk = 0; kk < BK; kk += 4) {
            const int kb = kk + 2 * hi;  // lanes 0-15: K={kk,kk+1}; 16-31: K={kk+2,kk+3}
            v2f a0 = *(const v2f*)(&As[(wm * 32 + ml     ) * LDA + kb]);
            v2f a1 = *(const v2f*)(&As[(wm * 32 + 16 + ml) * LDA + kb]);
            v2f b0, b1;
            b0.x = Bs[ kb      * LDB + wn * 32 + ml];
            b0.y = Bs[(kb + 1) * LDB + wn * 32 + ml];
            b1.x = Bs[ kb      * LDB + wn * 32 + 16 + ml];
            b1.y = Bs[(kb + 1) * LDB + wn * 32 + 16 + ml];
            acc[0][0] = __builtin_amdgcn_wmma_f32_16x16x4_f32(false, a0, false, b0, (short)0, acc[0][0], false, false);
            acc[0][1] = __builtin_amdgcn_wmma_f32_16x16x4_f32(false, a0, false, b1, (short)0, acc[0][1], false, false);
            acc[1][0] = __builtin_amdgcn_wmma_f32_16x16x4_f32(false, a1, false, b0, (short)0, acc[1][0], false, false);
            acc[1][1] = __builtin_amdgcn_wmma_f32_16x16x4_f32(false, a1, false, b1, (short)0, acc[1][1], false, false);
        }
        __syncthreads();
    }

    // Epilogue: masked hinge on accumulator fragments.
    // C/D layout: VGPR j -> M = j + 8*hi, N = ml.
    float part = 0.0f;
    #pragma unroll
    for (int tm = 0; tm < 2; tm++) {
        #pragma unroll
        for (int j = 0; j < 8; j++) {
            int   row = row0 + wm * 32 + tm * 16 + hi * 8 + j;
            float g   = gt[row];
            int   lb  = label[row];
            #pragma unroll
            for (int tn = 0; tn < 2; tn++) {
                int  nl    = wn * 32 + tn * 16 + ml;
                bool colok = (col0 + nl) < C;
                int  tcv   = tcs[nl];
                float h = 1.0f + acc[tm][tn][j] - g;
                if (colok && (tcv != lb) && (h > 0.0f)) part += h;
            }
        }
    }

    // Wave reduce -> LDS reduce -> one f32 atomic per block
    #pragma unroll
    for (int off = 16; off > 0; off >>= 1)
        part += __shfl_xor(part, off, 32);
    if (lane == 0) red[wid] = part;
    __syncthreads();
    if (tid == 0) {
        float s = 0.0f;
        #pragma unroll
        for (int i = 0; i < 8; i++) s += red[i];
        __hip_atomic_fetch_add(out, s, __ATOMIC_RELAXED, __HIP_MEMORY_SCOPE_AGENT);
    }
}

// ---------------------------------------------------------------------------
extern "C" void kernel_launch(void* const* d_in, const int* in_sizes, int n_in,
                              void* d_out, int out_size, void* d_ws, size_t ws_size,
                              hipStream_t stream) {
    const float* pred  = (const float*)d_in[0];
    const int*   label = (const int*)d_in[1];
    const int*   tcls  = (const int*)d_in[2];
    const float* sig   = (const float*)d_in[3];
    float*       out   = (float*)d_out;

    const int Bsz = in_sizes[1];           // 16384
    const int C   = in_sizes[2];           // 2000
    const int D   = in_sizes[0] / Bsz;     // 256

    float* gt = (float*)d_ws;              // Bsz floats of scratch

    rankloss_gt_kernel<<<Bsz / 8, 256, 0, stream>>>(pred, label, sig, gt, out, D, C);

    dim3 grid((C + BN - 1) / BN, Bsz / BM);
    rankloss_gemm_kernel<<<grid, 256, 0, stream>>>(pred, sig, tcls, label, gt, out, C, D);
}